// MatchingNetwork_67954972557334
// MI455X (gfx1250) — compile-verified
//
#include <hip/hip_runtime.h>
#include <math.h>

#define FEAT 512
#define NS   4096
#define NQ   8192
#define NCLS 64
#define KIT  3

typedef __attribute__((ext_vector_type(2))) float v2f;
typedef __attribute__((ext_vector_type(8))) float v8f;

__device__ __forceinline__ float sigm_f(float x) { return 1.f / (1.f + __expf(-x)); }

// ============================================================================
// WMMA fp32 GEMM:  C[M,N] = A[M,K] * op(B)  (+ C if accum, + bias if given)
// TRANS_B=true : B is [N,K] row-major (C = A B^T)
// TRANS_B=false: B is [K,N] row-major (C = A B)
// Register-blocked: one wave owns a 32x32 macro-tile = 2x2 WMMA accumulators,
// so each k-step is 2 A-frag + 2 B-frag loads feeding 4 V_WMMA_F32_16X16X4_F32
// (1 fragment load per WMMA). 8 waves/block arranged 2x4 -> 64x128 block tile;
// column-waves share A rows via L0/L2. M % 64 == 0, N % 32 == 0, K % 4 == 0.
// ============================================================================
template<bool TRANS_B>
__global__ __launch_bounds__(256) void gemm_wmma_f32(
    const float* __restrict__ A, const float* __restrict__ B,
    float* __restrict__ C, const float* __restrict__ bias,
    int N, int K, int lda, int ldb, int ldc, int accum)
{
  const int wave = threadIdx.x >> 5;
  const int lane = threadIdx.x & 31;
  const int t    = lane & 15;
  const int half = lane >> 4;
  const int wrow = wave >> 2;          // 0..1
  const int wcol = wave & 3;           // 0..3
  const int j0 = blockIdx.x * 128 + wcol * 32;
  const int i0 = blockIdx.y * 64 + wrow * 32;
  if (j0 >= N) return;

  v8f acc00, acc01, acc10, acc11;
  if (accum) {
#pragma unroll
    for (int r = 0; r < 8; ++r) {
      const int row0 = (i0 + 8 * half + r) * ldc;
      const int row1 = (i0 + 16 + 8 * half + r) * ldc;
      acc00[r] = C[row0 + j0 + t];
      acc01[r] = C[row0 + j0 + 16 + t];
      acc10[r] = C[row1 + j0 + t];
      acc11[r] = C[row1 + j0 + 16 + t];
    }
  } else if (bias) {
    float b0 = bias[j0 + t], b1 = bias[j0 + 16 + t];
#pragma unroll
    for (int r = 0; r < 8; ++r) {
      acc00[r] = b0; acc01[r] = b1; acc10[r] = b0; acc11[r] = b1;
    }
  } else {
#pragma unroll
    for (int r = 0; r < 8; ++r) {
      acc00[r] = 0.f; acc01[r] = 0.f; acc10[r] = 0.f; acc11[r] = 0.f;
    }
  }

  // A frag: lane (half,t) holds A[i+t][k + 2*half .. +1]   (16x4 tile, 2 VGPR)
  const float* Ap0 = A + (i0 + t) * lda + 2 * half;
  const float* Ap1 = Ap0 + 16 * lda;
  const float* Bt0 = B + (j0 + t) * ldb + 2 * half;   // used when TRANS_B
  const float* Bt1 = Bt0 + 16 * ldb;

#pragma unroll 4
  for (int k = 0; k < K; k += 4) {
    v2f a0 = *(const v2f*)(Ap0 + k);
    v2f a1 = *(const v2f*)(Ap1 + k);
    v2f b0, b1;
    if (TRANS_B) {
      b0 = *(const v2f*)(Bt0 + k);                    // B[j0+t   ][k+2h..+1]
      b1 = *(const v2f*)(Bt1 + k);                    // B[j0+16+t][k+2h..+1]
    } else {
      const float* bp = B + (k + 2 * half) * ldb + j0 + t;
      b0.x = bp[0];        b0.y = bp[ldb];            // B[k+2h..+1][j0+t]
      b1.x = bp[16];       b1.y = bp[ldb + 16];       // B[k+2h..+1][j0+16+t]
    }
    acc00 = __builtin_amdgcn_wmma_f32_16x16x4_f32(false, a0, false, b0, (short)0, acc00, false, false);
    acc01 = __builtin_amdgcn_wmma_f32_16x16x4_f32(false, a0, false, b1, (short)0, acc01, false, false);
    acc10 = __builtin_amdgcn_wmma_f32_16x16x4_f32(false, a1, false, b0, (short)0, acc10, false, false);
    acc11 = __builtin_amdgcn_wmma_f32_16x16x4_f32(false, a1, false, b1, (short)0, acc11, false, false);
  }

#pragma unroll
  for (int r = 0; r < 8; ++r) {
    const int row0 = (i0 + 8 * half + r) * ldc;
    const int row1 = (i0 + 16 + 8 * half + r) * ldc;
    C[row0 + j0 + t]      = acc00[r];
    C[row0 + j0 + 16 + t] = acc01[r];
    C[row1 + j0 + t]      = acc10[r];
    C[row1 + j0 + 16 + t] = acc11[r];
  }
}

// ============================================================================
// Serial bidirectional LSTM recurrence, input projections precomputed (X).
// 32 blocks per direction; block b owns hidden dims [16b,16b+16) => 64 gate
// rows, whose W_hh slice (128KB) lives LDS-resident in [K][64] layout
// (bank-conflict-free reads). h exchanged via L2 with a monotonic atomic
// step barrier, double-buffered.
// ============================================================================
__global__ __launch_bounds__(256) void lstm_scan_kernel(
    const float* __restrict__ Xf, const float* __restrict__ Xb,
    const float* __restrict__ Wf, const float* __restrict__ Wb,
    float* __restrict__ hsf, float* __restrict__ hsb,
    float* __restrict__ hglob, unsigned* __restrict__ cnts)
{
  __shared__ float WT[512 * 64];   // 128 KB: WT[k*64 + r] = W[grow(r)][k]
  __shared__ float hsm[512];
  __shared__ float psum[256];
  __shared__ float gatev[64];
  __shared__ float cst[16];

  const int dir = blockIdx.x >> 5;
  const int b   = blockIdx.x & 31;
  const int tid = threadIdx.x;
  const int d0  = b * 16;

  const float* X  = dir ? Xb : Xf;
  const float* W  = dir ? Wb : Wf;
  float* hs_out   = dir ? hsb : hsf;
  float* hg       = hglob + dir * 1024;            // [2][512]
  volatile unsigned* cnt = cnts + dir * 16;

  // stage W slice into LDS (rows: [i 0..15][f 16..31][g 32..47][o 48..63])
  for (int idx = tid; idx < 64 * 512; idx += 256) {
    int r = idx >> 9, k = idx & 511;
    int grow = (r >> 4) * 512 + d0 + (r & 15);
    WT[k * 64 + r] = W[grow * 512 + k];
  }
  for (int i = tid; i < 512; i += 256) hsm[i] = 0.f;
  if (tid < 16) cst[tid] = 0.f;
  __syncthreads();

  const int r    = tid & 63;
  const int part = tid >> 6;
  const int grow_t = (tid >> 4) * 512 + d0 + (tid & 15); // valid for tid<64

  for (int s = 0; s < 4096; ++s) {
    const int tstep = dir ? (4095 - s) : s;

    // 64 gate dots over K=512, 4 partial threads per row
    float sum = 0.f;
    const int kb = part * 128;
#pragma unroll 4
    for (int kk = 0; kk < 128; ++kk)
      sum += WT[(kb + kk) * 64 + r] * hsm[kb + kk];
    psum[tid] = sum;
    __syncthreads();

    if (tid < 64)
      gatev[tid] = psum[tid] + psum[tid + 64] + psum[tid + 128] +
                   psum[tid + 192] + X[tstep * 2048 + grow_t];
    __syncthreads();

    if (tid < 16) {
      float ig = sigm_f(gatev[tid]);
      float fg = sigm_f(gatev[16 + tid]);
      float gg = tanhf(gatev[32 + tid]);
      float og = sigm_f(gatev[48 + tid]);
      float cn = fg * cst[tid] + ig * gg;
      cst[tid] = cn;
      float hn = og * tanhf(cn);
      hs_out[tstep * 512 + d0 + tid]     = hn;
      hg[((s + 1) & 1) * 512 + d0 + tid] = hn;
    }
    __threadfence();
    __syncthreads();
    if (tid == 0) {
      atomicAdd((unsigned*)cnt, 1u);
      const unsigned target = 32u * (unsigned)(s + 1);
      while (*cnt < target) __builtin_amdgcn_s_sleep(2);
    }
    __syncthreads();
    __threadfence();
    if (s < 4095) {
      for (int i = tid; i < 512; i += 256)
        hsm[i] = hg[((s + 1) & 1) * 512 + i];
      __syncthreads();
    }
  }
}

// G = S + hf + hb ; Gn = G / (||G||_row + 1e-5)
__global__ __launch_bounds__(128) void build_G_kernel(
    const float* __restrict__ S, const float* __restrict__ hf,
    const float* __restrict__ hb, float* __restrict__ G, float* __restrict__ Gn)
{
  __shared__ float red[128];
  const int row = blockIdx.x, tid = threadIdx.x;
  float v[4], ss = 0.f;
#pragma unroll
  for (int e = 0; e < 4; ++e) {
    int idx = row * FEAT + tid + 128 * e;
    float x = S[idx] + hf[idx] + hb[idx];
    v[e] = x; G[idx] = x; ss += x * x;
  }
  red[tid] = ss;
  __syncthreads();
  for (int off = 64; off > 0; off >>= 1) {
    if (tid < off) red[tid] += red[tid + off];
    __syncthreads();
  }
  float inv = 1.f / (sqrtf(red[0]) + 1e-5f);
#pragma unroll
  for (int e = 0; e < 4; ++e)
    Gn[row * FEAT + tid + 128 * e] = v[e] * inv;
}

// in-place softmax over axis 0 (columns); one thread per column, online 2-pass
__global__ void softmax_axis0(float* __restrict__ X, int R, int C)
{
  int j = blockIdx.x * blockDim.x + threadIdx.x;
  if (j >= C) return;
  float m = -3.4e38f, s = 0.f;
  for (int i = 0; i < R; ++i) {
    float x = X[i * C + j];
    if (x > m) { s = s * __expf(m - x) + 1.f; m = x; }
    else       { s += __expf(x - m); }
  }
  float inv = 1.f / s;
  for (int i = 0; i < R; ++i)
    X[i * C + j] = __expf(X[i * C + j] - m) * inv;
}

// FCE LSTMCell elementwise + residual (h += f); gates row layout (i,f,g,o)
__global__ void fce_cell_kernel(const float* __restrict__ gates,
                                const float* __restrict__ fq,
                                float* __restrict__ h, float* __restrict__ c,
                                int n)
{
  int idx = blockIdx.x * blockDim.x + threadIdx.x;
  if (idx >= n) return;
  int row = idx >> 9, d = idx & 511;
  const float* g = gates + row * 2048;
  float ig = sigm_f(g[d]);
  float fg = sigm_f(g[512 + d]);
  float gt = tanhf(g[1024 + d]);
  float og = sigm_f(g[1536 + d]);
  float cn = fg * c[idx] + ig * gt;
  c[idx] = cn;
  h[idx] = og * tanhf(cn) + fq[idx];
}

__global__ void init_state_kernel(const float* __restrict__ fq,
                                  float* __restrict__ h, float* __restrict__ c,
                                  float* __restrict__ hglob,
                                  unsigned* __restrict__ cnts, int n)
{
  int idx = blockIdx.x * blockDim.x + threadIdx.x;
  if (idx < n) { h[idx] = fq[idx]; c[idx] = 0.f; }
  if (idx < 2048) hglob[idx] = 0.f;
  if (idx < 32)   cnts[idx]  = 0u;
}

__global__ void log_kernel(float* __restrict__ o, int n)
{
  int idx = blockIdx.x * blockDim.x + threadIdx.x;
  if (idx < n) o[idx] = logf(o[idx]);
}

// ============================================================================
extern "C" void kernel_launch(void* const* d_in, const int* in_sizes, int n_in,
                              void* d_out, int out_size, void* d_ws, size_t ws_size,
                              hipStream_t stream)
{
  const float* S     = (const float*)d_in[0];
  const float* fq    = (const float*)d_in[1];
  const float* Y_S   = (const float*)d_in[2];
  const float* Wf_ih = (const float*)d_in[3];
  const float* Wf_hh = (const float*)d_in[4];
  const float* bf    = (const float*)d_in[5];
  const float* Wb_ih = (const float*)d_in[6];
  const float* Wb_hh = (const float*)d_in[7];
  const float* bb    = (const float*)d_in[8];
  const float* Wc_ih = (const float*)d_in[9];
  const float* Wc_hh = (const float*)d_in[10];
  const float* bc    = (const float*)d_in[11];
  float* out = (float*)d_out;

  // workspace carve-out
  float* p = (float*)d_ws;
  auto carve = [&](size_t nf) { float* q = p; p += nf; return q; };
  float* Xf    = carve((size_t)NS * 2048);
  float* Xb    = carve((size_t)NS * 2048);
  float* hsf   = carve((size_t)NS * FEAT);
  float* hsb   = carve((size_t)NS * FEAT);
  float* G     = carve((size_t)NS * FEAT);
  float* Gn    = carve((size_t)NS * FEAT);
  float* h     = carve((size_t)NQ * FEAT);
  float* c     = carve((size_t)NQ * FEAT);
  float* rbuf  = carve((size_t)NQ * FEAT);
  float* gates = carve((size_t)NQ * 2048);
  float* Abig  = carve((size_t)NQ * NS);
  float* hglob = carve(2048);
  unsigned* cnts = (unsigned*)carve(64);

  auto gemm_nt = [&](const float* A, const float* B, float* C, const float* bias,
                     int M, int N, int K, int lda, int ldb, int ldc, int accum) {
    dim3 g((N + 127) / 128, M / 64);
    gemm_wmma_f32<true><<<g, dim3(256), 0, stream>>>(A, B, C, bias, N, K, lda, ldb, ldc, accum);
  };
  auto gemm_nn = [&](const float* A, const float* B, float* C, const float* bias,
                     int M, int N, int K, int lda, int ldb, int ldc, int accum) {
    dim3 g((N + 127) / 128, M / 64);
    gemm_wmma_f32<false><<<g, dim3(256), 0, stream>>>(A, B, C, bias, N, K, lda, ldb, ldc, accum);
  };

  const int nqf = NQ * FEAT;
  init_state_kernel<<<(nqf + 255) / 256, 256, 0, stream>>>(fq, h, c, hglob, cnts, nqf);

  // input projections for both LSTM directions (WMMA), bias folded in
  gemm_nt(S, Wf_ih, Xf, bf, NS, 2048, FEAT, FEAT, FEAT, 2048, 0);
  gemm_nt(S, Wb_ih, Xb, bb, NS, 2048, FEAT, FEAT, FEAT, 2048, 0);

  // serial recurrences: 32 blocks/direction, LDS-resident weight slices
  lstm_scan_kernel<<<64, 256, 0, stream>>>(Xf, Xb, Wf_hh, Wb_hh, hsf, hsb, hglob, cnts);

  build_G_kernel<<<NS, 128, 0, stream>>>(S, hsf, hsb, G, Gn);

  for (int it = 0; it < KIT; ++it) {
    gemm_nt(h, G, Abig, nullptr, NQ, NS, FEAT, FEAT, FEAT, NS, 0);     // logits = h @ G^T
    softmax_axis0<<<(NS + 255) / 256, 256, 0, stream>>>(Abig, NQ, NS); // softmax dim=0
    gemm_nn(Abig, G, rbuf, nullptr, NQ, FEAT, NS, NS, FEAT, FEAT, 0);  // r = a @ G
    // gates = [f|r] @ Wc_ih^T + bc + h @ Wc_hh^T  (split-K concat trick)
    gemm_nt(fq,   Wc_ih,       gates, bc,      NQ, 2048, FEAT, FEAT, 1024, 2048, 0);
    gemm_nt(rbuf, Wc_ih + 512, gates, nullptr, NQ, 2048, FEAT, FEAT, 1024, 2048, 1);
    gemm_nt(h,    Wc_hh,       gates, nullptr, NQ, 2048, FEAT, FEAT, FEAT, 2048, 1);
    fce_cell_kernel<<<(nqf + 255) / 256, 256, 0, stream>>>(gates, fq, h, c, nqf);
  }

  // scores -> column softmax -> class probs -> log
  gemm_nt(h, Gn, Abig, nullptr, NQ, NS, FEAT, FEAT, FEAT, NS, 0);
  softmax_axis0<<<(NS + 255) / 256, 256, 0, stream>>>(Abig, NQ, NS);
  gemm_nn(Abig, Y_S, out, nullptr, NQ, NCLS, NS, NS, NCLS, NCLS, 0);
  log_kernel<<<(NQ * NCLS + 255) / 256, 256, 0, stream>>>(out, NQ * NCLS);
}